// GraphClassifier_64750926954630
// MI455X (gfx1250) — compile-verified
//
#include <hip/hip_runtime.h>
#include <hip/hip_bf16.h>

namespace {

constexpr int NN = 2048;   // nodes
constexpr int NE = 4096;   // edges
constexpr int NB = 64;     // target pairs
constexpr int D  = 32;     // feature dim
constexpr float SLOPE = 0.2f;
constexpr int KSPLIT = 16; // K-partitioning for stage-1 WMMA

typedef __attribute__((ext_vector_type(2))) float v2f;
typedef __attribute__((ext_vector_type(8))) float v8f;

__device__ __forceinline__ float wredsum(float v) {
  v += __shfl_xor(v, 16, 32);
  v += __shfl_xor(v, 8, 32);
  v += __shfl_xor(v, 4, 32);
  v += __shfl_xor(v, 2, 32);
  v += __shfl_xor(v, 1, 32);
  return v;
}

// ---------------- setup ----------------
__global__ void k_init(int* owner, int* ocnt, int* icnt, int* ocur, int* icur) {
  int i = blockIdx.x * blockDim.x + threadIdx.x;
  if (i < NN) { owner[i] = -1; ocnt[i] = 0; icnt[i] = 0; ocur[i] = 0; icur[i] = 0; }
}

__global__ void k_owner(const int* head, const int* tail, int* owner) {
  int i = threadIdx.x;
  if (i < NB) {
    atomicMax(&owner[head[i]], i);
    atomicMax(&owner[tail[i]], i);
  }
}

__global__ void k_edge(const int* src, const int* dst, const int* rlab,
                       const int* owner, int* nbrel, int* ocnt, int* icnt) {
  int e = blockIdx.x * blockDim.x + threadIdx.x;
  if (e >= NE) return;
  int s = src[e], d = dst[e];
  int eo = max(owner[s], owner[d]);
  nbrel[e] = (eo >= 0) ? rlab[eo] : -1;       // neighbor flag + its relation label
  atomicAdd(&ocnt[s], 1);
  atomicAdd(&icnt[d], 1);
}

// single-block exclusive scan (Hillis-Steele over 2048, done twice)
__global__ void k_scan(const int* cntA, int* offA, const int* cntB, int* offB) {
  __shared__ int s[NN];
  int t = threadIdx.x;                        // 1024 threads
  for (int pass = 0; pass < 2; ++pass) {
    const int* cnt = pass ? cntB : cntA;
    int* off = pass ? offB : offA;
    s[t] = cnt[t]; s[t + 1024] = cnt[t + 1024];
    __syncthreads();
    for (int o = 1; o < NN; o <<= 1) {
      int i0 = t, i1 = t + 1024;
      int v0 = s[i0] + (i0 >= o ? s[i0 - o] : 0);
      int v1 = s[i1] + (i1 >= o ? s[i1 - o] : 0);
      __syncthreads();
      s[i0] = v0; s[i1] = v1;
      __syncthreads();
    }
    off[t] = s[t] - cnt[t];
    off[t + 1024] = s[t + 1024] - cnt[t + 1024];
    __syncthreads();
  }
}

__global__ void k_scatter(const int* src, const int* dst,
                          const int* ooff, int* ocur, int* olist,
                          const int* ioff, int* icur, int* ilist) {
  int e = blockIdx.x * blockDim.x + threadIdx.x;
  if (e >= NE) return;
  int s = src[e], d = dst[e];
  int po = atomicAdd(&ocur[s], 1);
  olist[ooff[s] + po] = e;
  int pi = atomicAdd(&icur[d], 1);
  ilist[ioff[d] + pi] = e;
}

__device__ void isort(int* a, int n) {
  for (int i = 1; i < n; ++i) {
    int key = a[i]; int j = i - 1;
    while (j >= 0 && a[j] > key) { a[j + 1] = a[j]; --j; }
    a[j + 1] = key;
  }
}

// sort each adjacency segment -> deterministic summation order across replays
__global__ void k_sort(int* olist, const int* ooff, const int* ocnt,
                       int* ilist, const int* ioff, const int* icnt) {
  int x = blockIdx.x * blockDim.x + threadIdx.x;
  if (x >= NN) return;
  isort(olist + ooff[x], ocnt[x]);
  isort(ilist + ioff[x], icnt[x]);
}

// ---------------- WMMA projection: proj[h] = X @ W[h]^T + b[h] ----------------
// X rows optionally gathered through rowmap (proj1 uses rel_emb_w[edge_type[e]]).
__global__ __launch_bounds__(32) void k_proj(const float* X, const int* rowmap,
                                             const float* W, const float* bias,
                                             float* out) {
  int l = threadIdx.x;
  int lm = l & 15, hi = l >> 4;
  int mt = blockIdx.x;            // 0..NE/16-1
  int h  = blockIdx.y;            // 0..5
  int row = mt * 16 + lm;
  int ar = rowmap ? rowmap[row] : row;
  const float* Xr = X + ar * D;
  const float* Wh = W + h * D * D;           // [f][d] row-major
  for (int nt = 0; nt < 2; ++nt) {
    int f = nt * 16 + lm;
    float bf = bias[h * D + f];
    v8f c;
#pragma unroll
    for (int r = 0; r < 8; ++r) c[r] = bf;   // C = bias, accumulate on top
#pragma unroll
    for (int k = 0; k < D; k += 4) {
      int ka = k + 2 * hi;
      v2f a, b;
      a.x = Xr[ka];             a.y = Xr[ka + 1];          // A 16x4 layout
      b.x = Wh[f * D + ka];     b.y = Wh[f * D + ka + 1];  // B 4x16 (= W^T col-read)
      c = __builtin_amdgcn_wmma_f32_16x16x4_f32(false, a, false, b, (short)0, c,
                                                false, false);
    }
#pragma unroll
    for (int r = 0; r < 8; ++r)
      out[(h * NE + mt * 16 + r + 8 * hi) * D + nt * 16 + lm] = c[r];
  }
}

// ---------------- stage-2 sparse masked-softmax attention ----------------
// one wave per edge-row; lane = feature dim. Unnormalized exp-sum then divide
// (exact: softmax is shift-free here since |logit| << 88).
__global__ void k_attn(const int* src, const int* dst, const int* etype,
                       const int* nbrel, const float* rel_emb,
                       const int* ooff, const int* ocnt, const int* olist,
                       const int* ioff, const int* icnt, const int* ilist,
                       const float* proj, float* h1) {
  int lane = threadIdx.x & 31;
  int e = blockIdx.x * (blockDim.x >> 5) + (threadIdx.x >> 5);
  if (e >= NE) return;
  int d = lane;
  float base = rel_emb[etype[e] * D + d];
  int rel = nbrel[e];
  if (rel < 0) { h1[e * D + d] = base; return; }   // non-neighbor rows: no h0N
  int u = src[e], v = dst[e];
  float qd = rel_emb[rel * D + d];
  // six (list, predicate) configs == the six dense masks:
  // 0: in(u)&src!=v  1: out(u)&dst!=v  2: in(v)&src!=u
  // 3: out(v)&dst!=u 4: out(u)&dst==v  5: in(u)&src==v
  int cfg_node[6] = {u, u, v, v, u, u};
  int cfg_in[6]   = {1, 0, 1, 0, 0, 1};
  int cfg_cmp[6]  = {v, v, u, u, v, v};
  int cfg_eq[6]   = {0, 0, 0, 0, 1, 1};
  float hacc = 0.f;
#pragma unroll
  for (int i = 0; i < 6; ++i) {
    int x0 = cfg_node[i];
    const int* off = cfg_in[i] ? ioff : ooff;
    const int* cnt = cfg_in[i] ? icnt : ocnt;
    const int* lst = cfg_in[i] ? ilist : olist;
    int s0 = off[x0], c0 = cnt[x0];
    float Z = 0.f, acc = 0.f;
    for (int j = 0; j < c0; ++j) {
      int n = lst[s0 + j];
      float hv = rel_emb[etype[n] * D + d];
      float x = wredsum(qd * hv);                      // logit (all lanes)
      x = (x >= 0.f) ? x : SLOPE * x;                  // LeakyReLU
      int fld = cfg_in[i] ? src[n] : dst[n];
      bool pred = cfg_eq[i] ? (fld == cfg_cmp[i]) : (fld != cfg_cmp[i]);
      if (pred && x != 0.f) {                          // vb = atten != 0
        float ex = expf(x);
        Z += ex;
        acc += ex * proj[(i * NE + n) * D + d];
      }
    }
    if (Z > 0.f) hacc += acc / Z;
  }
  h1[e * D + d] = base + fmaxf(hacc, 0.f);             // relu(h0N) added
}

// ---------------- stage-1 WMMA: part[z] += sum_i mask_i[b,n] * proj2[i][n,:] ----
// A (0/1 masks) generated on the fly directly in WMMA A-layout.
__global__ __launch_bounds__(32) void k_stage1(const int* src, const int* dst,
                                               const int* head, const int* tail,
                                               const float* proj, float* part) {
  int l = threadIdx.x;
  int lm = l & 15, hi = l >> 4;
  int b0 = blockIdx.x * 16;      // 4 M-tiles
  int f0 = blockIdx.y * 16;      // 2 N-tiles
  int z  = blockIdx.z;           // K split
  int u = head[b0 + lm], v = tail[b0 + lm];
  v8f c;
#pragma unroll
  for (int r = 0; r < 8; ++r) c[r] = 0.f;
  int n_beg = z * (NE / KSPLIT);
  int n_end = n_beg + (NE / KSPLIT);
  for (int k = n_beg; k < n_end; k += 4) {
    int na = k + 2 * hi;                                // this lane's two K cols
    __builtin_prefetch(&proj[(5 * NE + na + 4) * D], 0, 1);
    int s0 = src[na], d0 = dst[na];
    int s1 = src[na + 1], d1 = dst[na + 1];
    bool ieo0 = (d0 == u), oeo0 = (s0 == u), iei0 = (d0 == v), oei0 = (s0 == v);
    bool ieo1 = (d1 == u), oeo1 = (s1 == u), iei1 = (d1 == v), oei1 = (s1 == v);
    float ax[6] = {(ieo0 && !oei0) ? 1.f : 0.f, (oeo0 && !iei0) ? 1.f : 0.f,
                   (iei0 && !oeo0) ? 1.f : 0.f, (oei0 && !ieo0) ? 1.f : 0.f,
                   (oeo0 && iei0) ? 1.f : 0.f,  (ieo0 && oei0) ? 1.f : 0.f};
    float ay[6] = {(ieo1 && !oei1) ? 1.f : 0.f, (oeo1 && !iei1) ? 1.f : 0.f,
                   (iei1 && !oeo1) ? 1.f : 0.f, (oei1 && !ieo1) ? 1.f : 0.f,
                   (oeo1 && iei1) ? 1.f : 0.f,  (ieo1 && oei1) ? 1.f : 0.f};
#pragma unroll
    for (int i = 0; i < 6; ++i) {
      v2f a; a.x = ax[i]; a.y = ay[i];
      const float* P = proj + (i * NE + na) * D + f0 + lm;
      v2f bb; bb.x = P[0]; bb.y = P[D];
      c = __builtin_amdgcn_wmma_f32_16x16x4_f32(false, a, false, bb, (short)0, c,
                                                false, false);
    }
  }
#pragma unroll
  for (int r = 0; r < 8; ++r)
    part[(z * NB + b0 + r + 8 * hi) * D + f0 + lm] = c[r];
}

// fixed-order reduction of K-split partials (deterministic, no float atomics)
__global__ void k_red(const float* part, float* h1N) {
  int j = blockIdx.x * blockDim.x + threadIdx.x;
  if (j >= NB * D) return;
  float s = 0.f;
  for (int z = 0; z < KSPLIT; ++z) s += part[z * NB * D + j];
  h1N[j] = s;
}

// h2 = h1[:B] + relu(h1N); normalize; score = g @ Wout + bout
__global__ void k_final(const float* h1, const float* h1N, const float* Wout,
                        const float* bout, float* out) {
  int b = blockIdx.x, d = threadIdx.x;
  float h2 = h1[b * D + d] + fmaxf(h1N[b * D + d], 0.f);
  float ss = wredsum(h2 * h2);
  float nrm = fmaxf(sqrtf(ss), 1e-12f);
  float g = h2 / nrm;
  float pr = wredsum(g * Wout[d]);
  if (d == 0) out[b] = pr + bout[0];
}

} // namespace

extern "C" void kernel_launch(void* const* d_in, const int* in_sizes, int n_in,
                              void* d_out, int out_size, void* d_ws, size_t ws_size,
                              hipStream_t stream) {
  const int* src   = (const int*)d_in[0];
  const int* dst   = (const int*)d_in[1];
  const int* etype = (const int*)d_in[2];
  const int* head  = (const int*)d_in[3];
  const int* tail  = (const int*)d_in[4];
  const int* rlab  = (const int*)d_in[5];
  // d_in[6] = num_nodes (device scalar, constant 2048 per reference)
  const float* rel_emb = (const float*)d_in[7];
  const float* W1  = (const float*)d_in[8];
  const float* b1  = (const float*)d_in[9];
  const float* W2  = (const float*)d_in[10];
  const float* b2  = (const float*)d_in[11];
  const float* Wout = (const float*)d_in[12];
  const float* bout = (const float*)d_in[13];
  float* out = (float*)d_out;

  char* p = (char*)d_ws;
  auto alloc = [&](size_t bytes) { char* r = p; p += (bytes + 255) & ~size_t(255); return r; };
  int* owner  = (int*)alloc(NN * 4);
  int* ocnt   = (int*)alloc(NN * 4);
  int* icnt   = (int*)alloc(NN * 4);
  int* ooff   = (int*)alloc(NN * 4);
  int* ioff   = (int*)alloc(NN * 4);
  int* ocur   = (int*)alloc(NN * 4);
  int* icur   = (int*)alloc(NN * 4);
  int* nbrel  = (int*)alloc(NE * 4);
  int* olist  = (int*)alloc(NE * 4);
  int* ilist  = (int*)alloc(NE * 4);
  float* h1   = (float*)alloc((size_t)NE * D * 4);
  float* proj = (float*)alloc((size_t)6 * NE * D * 4);   // reused for proj1 then proj2
  float* part = (float*)alloc((size_t)KSPLIT * NB * D * 4);
  float* h1N  = (float*)alloc((size_t)NB * D * 4);
  (void)ws_size; (void)n_in; (void)in_sizes; (void)out_size;

  k_init   <<<(NN + 255) / 256, 256, 0, stream>>>(owner, ocnt, icnt, ocur, icur);
  k_owner  <<<1, 64, 0, stream>>>(head, tail, owner);
  k_edge   <<<NE / 256, 256, 0, stream>>>(src, dst, rlab, owner, nbrel, ocnt, icnt);
  k_scan   <<<1, 1024, 0, stream>>>(ocnt, ooff, icnt, ioff);
  k_scatter<<<NE / 256, 256, 0, stream>>>(src, dst, ooff, ocur, olist, ioff, icur, ilist);
  k_sort   <<<NN / 256, 256, 0, stream>>>(olist, ooff, ocnt, ilist, ioff, icnt);

  // proj1[i] = rel_emb[edge_type] @ W1[i]^T + b1[i]  (WMMA f32 16x16x4)
  k_proj   <<<dim3(NE / 16, 6), 32, 0, stream>>>(rel_emb, etype, W1, b1, proj);
  // stage-2 sparse masked softmax -> h1
  k_attn   <<<NE / 8, 256, 0, stream>>>(src, dst, etype, nbrel, rel_emb,
                                        ooff, ocnt, olist, ioff, icnt, ilist, proj, h1);
  // proj2[i] = h1 @ W2[i]^T + b2[i]
  k_proj   <<<dim3(NE / 16, 6), 32, 0, stream>>>(h1, nullptr, W2, b2, proj);
  // stage-1 mask matmul (WMMA, on-the-fly 0/1 A-operand), K split 16 ways
  k_stage1 <<<dim3(NB / 16, 2, KSPLIT), 32, 0, stream>>>(src, dst, head, tail, proj, part);
  k_red    <<<(NB * D + 255) / 256, 256, 0, stream>>>(part, h1N);
  k_final  <<<NB, 32, 0, stream>>>(h1, h1N, Wout, bout, out);
}